// MaskedChamferLoss_64080912056612
// MI455X (gfx1250) — compile-verified
//
#include <hip/hip_runtime.h>

// MaskedChamferLoss for MI455X (gfx1250, wave32).
// B=16, L=1024, F=6. p in {1,2}.
// p=2 distances via V_WMMA_F32_16X16X4_F32 (Gram matrix trick),
// p=1 distances on VALU (co-executes with WMMA per ISA 8.4),
// mins via ds_min_u32 on float-bits, means via ds_add_f32.

typedef float v2f __attribute__((ext_vector_type(2)));
typedef float v8f __attribute__((ext_vector_type(8)));

#define B_DIM   16
#define L_DIM   1024
#define F_DIM   6
#define FP      8                      // K padded to 8 for two K=4 WMMAs
#define TILES1D (L_DIM / 16)           // 64
#define NTILES  (TILES1D * TILES1D)    // 4096
#define INF_BITS 0x7F800000u

__global__ __launch_bounds__(1024)
void chamfer_batch_kernel(const float* __restrict__ xp,
                          const float* __restrict__ xt,
                          const unsigned char* __restrict__ mask,
                          float* __restrict__ partial)
{
    const int b   = blockIdx.x;
    const int tid = threadIdx.x;

    extern __shared__ char smem_raw[];
    float*    sXp    = (float*)smem_raw;            // [L][FP]  32 KB
    float*    sXt    = sXp + L_DIM * FP;            // [L][FP]  32 KB
    float*    sNp    = sXt + L_DIM * FP;            // [L]       4 KB
    float*    sNt    = sNp + L_DIM;                 // [L]       4 KB
    unsigned* sMinI1 = (unsigned*)(sNt + L_DIM);    // min over masked i, p=1
    unsigned* sMinI2 = sMinI1 + L_DIM;              // min over masked i, p=2
    unsigned* sMinJ1 = sMinI2 + L_DIM;              // min over masked j, p=1
    unsigned* sMinJ2 = sMinJ1 + L_DIM;              // min over masked j, p=2
    unsigned char* sMask = (unsigned char*)(sMinJ2 + L_DIM); // [L] 1 KB

    __shared__ float acc[5];  // sumI1, sumI2, sumJ1, sumJ2, cnt

    const float* xpb = xp + (size_t)b * L_DIM * F_DIM;
    const float* xtb = xt + (size_t)b * L_DIM * F_DIM;
    const unsigned char* mb = mask + (size_t)b * L_DIM;

    // ---- stage batch into LDS (padded K stride), init state ----
    for (int idx = tid; idx < L_DIM * F_DIM; idx += blockDim.x) {
        const int row = idx / F_DIM;
        const int col = idx - row * F_DIM;
        sXp[row * FP + col] = xpb[idx];
        sXt[row * FP + col] = xtb[idx];
    }
    for (int l = tid; l < L_DIM; l += blockDim.x) {
        sXp[l * FP + 6] = 0.f; sXp[l * FP + 7] = 0.f;
        sXt[l * FP + 6] = 0.f; sXt[l * FP + 7] = 0.f;
        sMask[l]  = mb[l];
        sMinI1[l] = INF_BITS; sMinI2[l] = INF_BITS;
        sMinJ1[l] = INF_BITS; sMinJ2[l] = INF_BITS;
    }
    if (tid < 5) acc[tid] = 0.f;
    __syncthreads();

    // ---- squared norms (for the Gram-matrix distance identity) ----
    for (int l = tid; l < L_DIM; l += blockDim.x) {
        float s = 0.f, t = 0.f;
        #pragma unroll
        for (int k = 0; k < F_DIM; ++k) {
            const float a = sXp[l * FP + k]; s += a * a;
            const float c = sXt[l * FP + k]; t += c * c;
        }
        sNp[l] = s;
        sNt[l] = t;
    }
    __syncthreads();

    // ---- 16x16 tile sweep: one wave per tile-pair, round-robin ----
    const int wave   = tid >> 5;
    const int lane   = tid & 31;
    const int half   = lane >> 4;    // 0: lanes 0-15, 1: lanes 16-31
    const int lm     = lane & 15;
    const int nWaves = blockDim.x >> 5;   // 32

    for (int t = wave; t < NTILES; t += nWaves) {
        const int i0 = (t >> 6) << 4;
        const int j0 = (t & (TILES1D - 1)) << 4;

        // A (16x4 f32): lane holds row M=lm, K = 2*half + {0,1}; chunk2 adds K+4.
        // B (4x16 f32): lane holds col N=lm, same K striping (symmetric layout).
        const int ar = (i0 + lm) * FP + half * 2;
        const int br = (j0 + lm) * FP + half * 2;
        v2f a0, a1, b0, b1;
        a0[0] = sXp[ar];     a0[1] = sXp[ar + 1];
        a1[0] = sXp[ar + 4]; a1[1] = sXp[ar + 5];
        b0[0] = sXt[br];     b0[1] = sXt[br + 1];
        b1[0] = sXt[br + 4]; b1[1] = sXt[br + 5];

        v8f c = {0.f, 0.f, 0.f, 0.f, 0.f, 0.f, 0.f, 0.f};
        c = __builtin_amdgcn_wmma_f32_16x16x4_f32(false, a0, false, b0,
                                                  (short)0, c, false, false);
        c = __builtin_amdgcn_wmma_f32_16x16x4_f32(false, a1, false, b1,
                                                  (short)0, c, false, false);

        // C/D layout: VGPR v, lane -> (M = v + 8*half, N = lm)
        const int  j     = j0 + lm;
        const bool maskJ = sMask[j] != 0;
        const float ntj  = sNt[j];
        float xtv[F_DIM];
        #pragma unroll
        for (int k = 0; k < F_DIM; ++k) xtv[k] = sXt[j * FP + k];

        unsigned colMin1 = INF_BITS, colMin2 = INF_BITS;
        #pragma unroll
        for (int v = 0; v < 8; ++v) {
            const int i = i0 + half * 8 + v;
            // p=1 on VALU (co-executes with the WMMA pipe)
            float d1 = 0.f;
            #pragma unroll
            for (int k = 0; k < F_DIM; ++k)
                d1 += fabsf(sXp[i * FP + k] - xtv[k]);
            // p=2 from WMMA Gram matrix
            const float d2 = fmaxf(sNp[i] + ntj - 2.f * c[v], 0.f);

            const unsigned u1 = __float_as_uint(d1);
            const unsigned u2 = __float_as_uint(d2);
            if (sMask[i]) {                     // min over masked i -> minI[j]
                colMin1 = (u1 < colMin1) ? u1 : colMin1;
                colMin2 = (u2 < colMin2) ? u2 : colMin2;
            }
            if (maskJ) {                        // min over masked j -> minJ[i]
                atomicMin(&sMinJ1[i], u1);
                atomicMin(&sMinJ2[i], u2);
            }
        }
        atomicMin(&sMinI1[j], colMin1);
        atomicMin(&sMinI2[j], colMin2);
    }
    __syncthreads();

    // ---- masked means via LDS float-add atomics ----
    for (int l = tid; l < L_DIM; l += blockDim.x) {
        if (sMask[l]) {
            atomicAdd(&acc[0], __uint_as_float(sMinI1[l]));
            atomicAdd(&acc[1], __uint_as_float(sMinI2[l]));
            atomicAdd(&acc[2], __uint_as_float(sMinJ1[l]));
            atomicAdd(&acc[3], __uint_as_float(sMinJ2[l]));
            atomicAdd(&acc[4], 1.f);
        }
    }
    __syncthreads();

    if (tid == 0) {
        const float cnt = acc[4];
        const float ch1 = 0.5f * (acc[0] + acc[2]) / cnt;  // chamfer p=1
        const float ch2 = 0.5f * (acc[1] + acc[3]) / cnt;  // chamfer p=2
        partial[b] = ch1 + ch2;
    }
}

__global__ void chamfer_finalize_kernel(const float* __restrict__ partial,
                                        float* __restrict__ out)
{
    if (threadIdx.x == 0 && blockIdx.x == 0) {
        float s = 0.f;
        #pragma unroll
        for (int b = 0; b < B_DIM; ++b) s += partial[b];
        // loss = mean over (2 p-values x 16 batches)
        out[0] = s / (2.0f * (float)B_DIM);
    }
}

extern "C" void kernel_launch(void* const* d_in, const int* in_sizes, int n_in,
                              void* d_out, int out_size, void* d_ws, size_t ws_size,
                              hipStream_t stream) {
    const float* xp = (const float*)d_in[0];             // (16,1024,6) f32
    const float* xt = (const float*)d_in[1];             // (16,1024,6) f32
    const unsigned char* mask = (const unsigned char*)d_in[2]; // (16,1024) bool
    float* out = (float*)d_out;          // [loss(1), x_pred(98304), x_true(98304)]
    float* partial = (float*)d_ws;       // 16 per-batch partial sums

    const size_t nElem = (size_t)B_DIM * L_DIM * F_DIM;  // 98304

    // LDS: 2*(1024*8) + 2*1024 floats + 4*1024 uints + 1024 bytes
    const size_t smemBytes =
        (2 * (size_t)L_DIM * FP + 2 * (size_t)L_DIM + 4 * (size_t)L_DIM) * 4
        + (size_t)L_DIM;

    chamfer_batch_kernel<<<dim3(B_DIM), dim3(1024), smemBytes, stream>>>(
        xp, xt, mask, partial);
    chamfer_finalize_kernel<<<dim3(1), dim3(32), 0, stream>>>(partial, out);

    // Tuple passthrough outputs: x_pred, x_true
    hipMemcpyAsync(out + 1,           xp, nElem * sizeof(float),
                   hipMemcpyDeviceToDevice, stream);
    hipMemcpyAsync(out + 1 + nElem,   xt, nElem * sizeof(float),
                   hipMemcpyDeviceToDevice, stream);
}